// FFM_19189913878982
// MI455X (gfx1250) — compile-verified
//
#include <hip/hip_runtime.h>

#define N_FEAT   2048
#define N_FIELD  32
#define N_FACTOR 16
#define BATCH    8192

typedef __attribute__((ext_vector_type(2))) float v2f;
typedef __attribute__((ext_vector_type(8))) float v8f;
typedef __attribute__((ext_vector_type(4))) int   v4i;

#if __has_builtin(__builtin_amdgcn_global_load_async_to_lds_b128)
#define HAVE_ASYNC_LDS 1
typedef __attribute__((address_space(1))) v4i* gv4i_p;   // global (AS1) int4*
typedef __attribute__((address_space(3))) v4i* lv4i_p;   // LDS (AS3) int4*
__device__ __forceinline__ void async_tile_b128(const float* g, float* l) {
    __builtin_amdgcn_global_load_async_to_lds_b128(
        (gv4i_p)(void*)(const_cast<float*>(g)), (lv4i_p)(void*)l, 0, 0);
}
__device__ __forceinline__ void wait_async0() {
#if __has_builtin(__builtin_amdgcn_s_wait_asynccnt)
    __builtin_amdgcn_s_wait_asynccnt(0);
#else
    asm volatile("s_wait_asynccnt 0x0" ::: "memory");
#endif
}
#else
#define HAVE_ASYNC_LDS 0
#endif

// ---------------------------------------------------------------------------
// Kernel 1: W[i][j] = dot(V[i, field[j], :], V[j, field[i], :]) over 16 factors
//           diag[i] = W[i][i]     (W is symmetric by construction)
// ---------------------------------------------------------------------------
__global__ __launch_bounds__(256)
void ffm_build_W(const float* __restrict__ V,
                 const int* __restrict__ field,
                 float* __restrict__ W,
                 float* __restrict__ diag) {
    const int j = blockIdx.x * 256 + threadIdx.x;
    const int i = blockIdx.y;
    const int fi = field[i];          // uniform per block -> scalar load
    const int fj = field[j];
    const float4* vi = (const float4*)(V + ((size_t)i * N_FIELD + fj) * N_FACTOR);
    const float4* vj = (const float4*)(V + ((size_t)j * N_FIELD + fi) * N_FACTOR);
    float s = 0.f;
#pragma unroll
    for (int q = 0; q < 4; ++q) {
        float4 a = vi[q], b = vj[q];
        s += a.x * b.x + a.y * b.y + a.z * b.z + a.w * b.w;
    }
    W[(size_t)i * N_FEAT + j] = s;
    if (i == j) diag[i] = s;
}

// ---------------------------------------------------------------------------
// Kernel 2: tiled GEMM y = x @ W via V_WMMA_F32_16X16X4_F32 with fused epilogue
//   partials[b, cb] = sum_{j in col-block cb} x[b,j]*(y[b,j] - x[b,j]*diag[j])
//
// Block tile 128x128, BK=16, 8 waves (2 row x 4 col), wave tile 64x32 (4x2 WMMA).
// Fragment-ready LDS layout [kgroup][row][k%4]: one ds_load_b64 per fragment.
// B tile uses W's symmetry so both tiles load coalesced row-major.
// Global->LDS staging uses GLOBAL_LOAD_ASYNC_TO_LDS_B128 (ASYNCcnt) when the
// builtin exists; otherwise reg staging pinned early via sched_barrier.
// ---------------------------------------------------------------------------
__global__ __launch_bounds__(256)
void ffm_gemm_wmma(const float* __restrict__ x,
                   const float* __restrict__ W,
                   const float* __restrict__ diag,
                   float* __restrict__ partials) {
    // [buf][kgroup(0..3)][row(128, padded to 132)][k-in-group(0..3)]
    __shared__ float As[2][4][132][4];
    __shared__ float Bs[2][4][132][4];
    __shared__ float red[4][128];      // per-wave-column row partials

    const int tid  = threadIdx.x;
    const int lane = tid & 31;
    const int wv   = tid >> 5;
    const int wr   = wv >> 2;          // 0..1  (wave row)
    const int wc   = wv & 3;           // 0..3  (wave col)
    const int half = lane >> 4;        // 0: lanes 0-15 (K 0/1), 1: lanes 16-31 (K 2/3)
    const int l16  = lane & 15;

    const int bM = blockIdx.y * 128;   // batch-row base
    const int bN = blockIdx.x * 128;   // feature-col base

    v8f acc[4][2];
    {
        v8f zero = {0.f, 0.f, 0.f, 0.f, 0.f, 0.f, 0.f, 0.f};
#pragma unroll
        for (int rt = 0; rt < 4; ++rt)
#pragma unroll
            for (int ct = 0; ct < 2; ++ct) acc[rt][ct] = zero;
    }

    // tile-loader work distribution: both tiles use the same pattern
    const int lq = tid >> 2;           // row (A) / col (B): lq and lq+64
    const int lc = tid & 3;            // k-group -> float4 at k0 + 4*lc

    const float* Ag0 = x + ((size_t)bM + lq) * N_FEAT + 4 * lc;
    const float* Ag1 = Ag0 + (size_t)64 * N_FEAT;
    const float* Bg0 = W + ((size_t)bN + lq) * N_FEAT + 4 * lc;   // symmetry
    const float* Bg1 = Bg0 + (size_t)64 * N_FEAT;

    const int KT = N_FEAT / 16;        // 128 K-tiles

#if HAVE_ASYNC_LDS
    // ---- prologue: async-copy tile 0 straight into LDS ----
    async_tile_b128(Ag0, &As[0][lc][lq][0]);
    async_tile_b128(Ag1, &As[0][lc][lq + 64][0]);
    async_tile_b128(Bg0, &Bs[0][lc][lq][0]);
    async_tile_b128(Bg1, &Bs[0][lc][lq + 64][0]);
    wait_async0();
    __syncthreads();
#else
    {
        float4 a0 = *(const float4*)Ag0;
        float4 a1 = *(const float4*)Ag1;
        float4 b0 = *(const float4*)Bg0;
        float4 b1 = *(const float4*)Bg1;
        *(float4*)&As[0][lc][lq][0]      = a0;
        *(float4*)&As[0][lc][lq + 64][0] = a1;
        *(float4*)&Bs[0][lc][lq][0]      = b0;
        *(float4*)&Bs[0][lc][lq + 64][0] = b1;
    }
    __syncthreads();
#endif

    for (int kt = 0; kt < KT; ++kt) {
        const int cur = kt & 1, nxt = cur ^ 1;
        const bool more = (kt + 1) < KT;
#if HAVE_ASYNC_LDS
        if (more) {
            // buffer nxt was retired by last barrier -> safe to overwrite;
            // transfers fly during the whole compute phase below (ASYNCcnt).
            const int k0 = (kt + 1) * 16;
            async_tile_b128(Ag0 + k0, &As[nxt][lc][lq][0]);
            async_tile_b128(Ag1 + k0, &As[nxt][lc][lq + 64][0]);
            async_tile_b128(Bg0 + k0, &Bs[nxt][lc][lq][0]);
            async_tile_b128(Bg1 + k0, &Bs[nxt][lc][lq + 64][0]);
        }
#else
        float4 a0, a1, b0, b1;
        if (more) {
            const int k0 = (kt + 1) * 16;
            a0 = *(const float4*)(Ag0 + k0);
            a1 = *(const float4*)(Ag1 + k0);
            b0 = *(const float4*)(Bg0 + k0);
            b1 = *(const float4*)(Bg1 + k0);
            if (kt + 2 < KT) {
                const int k2 = (kt + 2) * 16;
                __builtin_prefetch(Ag0 + k2, 0, 1);
                __builtin_prefetch(Bg0 + k2, 0, 1);
            }
            __builtin_amdgcn_sched_barrier(0);   // keep loads in flight over WMMAs
        }
#endif
        // ---- compute on current buffer: 4 k-substeps of 4, 4x2 WMMA tiles ----
#pragma unroll
        for (int kk = 0; kk < 4; ++kk) {
            v2f afr[4], bfr[2];
#pragma unroll
            for (int rt = 0; rt < 4; ++rt) {
                const int r = wr * 64 + rt * 16 + l16;
                afr[rt] = *(const v2f*)&As[cur][kk][r][half * 2];   // ds_load_b64
            }
#pragma unroll
            for (int ct = 0; ct < 2; ++ct) {
                const int c = wc * 32 + ct * 16 + l16;
                bfr[ct] = *(const v2f*)&Bs[cur][kk][c][half * 2];   // ds_load_b64
            }
#pragma unroll
            for (int rt = 0; rt < 4; ++rt)
#pragma unroll
                for (int ct = 0; ct < 2; ++ct)
                    acc[rt][ct] = __builtin_amdgcn_wmma_f32_16x16x4_f32(
                        false, afr[rt], false, bfr[ct],
                        (short)0, acc[rt][ct], false, false);
        }
#if HAVE_ASYNC_LDS
        if (more) wait_async0();   // this wave's writes into buffer nxt landed
#else
        if (more) {
            *(float4*)&As[nxt][lc][lq][0]      = a0;
            *(float4*)&As[nxt][lc][lq + 64][0] = a1;
            *(float4*)&Bs[nxt][lc][lq][0]      = b0;
            *(float4*)&Bs[nxt][lc][lq + 64][0] = b1;
        }
#endif
        __syncthreads();
    }

    // ---- epilogue: per-row partial of x*(y - x*diag), deterministic shuffles ----
    // C layout: VGPR i -> row (tile + i + half*8), col (tile + l16)
#pragma unroll
    for (int rt = 0; rt < 4; ++rt) {
#pragma unroll
        for (int i = 0; i < 8; ++i) {
            const int lrow = wr * 64 + rt * 16 + i + half * 8;
            const int row  = bM + lrow;
            float s = 0.f;
#pragma unroll
            for (int ct = 0; ct < 2; ++ct) {
                const int col = bN + wc * 32 + ct * 16 + l16;
                const float xv = x[(size_t)row * N_FEAT + col];
                float cval = xv * (acc[rt][ct][i] - xv * diag[col]);
                // reduce across the 16 lanes of this half (masks < 16 stay in-half)
                cval += __shfl_xor(cval, 1, 32);
                cval += __shfl_xor(cval, 2, 32);
                cval += __shfl_xor(cval, 4, 32);
                cval += __shfl_xor(cval, 8, 32);
                s += cval;
            }
            if (l16 == 0) red[wc][lrow] = s;   // lanes 0 and 16 own distinct rows
        }
    }
    __syncthreads();
    if (tid < 128) {
        const float p = red[0][tid] + red[1][tid] + red[2][tid] + red[3][tid];
        partials[(size_t)(bM + tid) * gridDim.x + blockIdx.x] = p;
    }
}

// ---------------------------------------------------------------------------
// Kernel 3: out[b] = x[b]·lin_w + lin_b + 0.5 * sum_cb partials[b, cb]
// ---------------------------------------------------------------------------
__global__ __launch_bounds__(256)
void ffm_finalize(const float* __restrict__ x,
                  const float* __restrict__ lin_w,
                  const float* __restrict__ lin_b,
                  const float* __restrict__ partials,
                  float* __restrict__ out,
                  int ncb) {
    __shared__ float sm[256];
    const int b = blockIdx.x;
    const int tid = threadIdx.x;
    float s = 0.f;
    for (int j = tid; j < N_FEAT; j += 256)
        s += x[(size_t)b * N_FEAT + j] * lin_w[j];
    if (tid < ncb) s += 0.5f * partials[(size_t)b * ncb + tid];
    sm[tid] = s;
    __syncthreads();
    for (int off = 128; off > 0; off >>= 1) {
        if (tid < off) sm[tid] += sm[tid + off];
        __syncthreads();
    }
    if (tid == 0) out[b] = sm[0] + lin_b[0];
}

// ---------------------------------------------------------------------------
extern "C" void kernel_launch(void* const* d_in, const int* in_sizes, int n_in,
                              void* d_out, int out_size, void* d_ws, size_t ws_size,
                              hipStream_t stream) {
    const float* x     = (const float*)d_in[0];
    const float* V     = (const float*)d_in[1];
    const float* lin_w = (const float*)d_in[2];
    const float* lin_b = (const float*)d_in[3];
    const int*   field = (const int*)d_in[4];

    float* W        = (float*)d_ws;                       // 2048*2048 f32 = 16 MB
    float* diag     = W + (size_t)N_FEAT * N_FEAT;        // 2048 f32
    float* partials = diag + N_FEAT;                      // 8192*16 f32 = 512 KB

    (void)in_sizes; (void)n_in; (void)out_size; (void)ws_size;

    ffm_build_W<<<dim3(N_FEAT / 256, N_FEAT), 256, 0, stream>>>(V, field, W, diag);
    ffm_gemm_wmma<<<dim3(N_FEAT / 128, BATCH / 128), 256, 0, stream>>>(x, W, diag, partials);
    ffm_finalize<<<BATCH, 256, 0, stream>>>(x, lin_w, lin_b, partials, (float*)d_out,
                                            N_FEAT / 128);
}